// Dyna_Dec_36670430773802
// MI455X (gfx1250) — compile-verified
//
#include <hip/hip_runtime.h>
#include <hip/hip_bf16.h>

// Out_l (128x32) = X_l (128x32) @ W_l (32x32) + bias_l, for L = 4096 positions.
// fp32 WMMA (v_wmma_f32_16x16x4_f32) keeps reference precision; AI ~ 7 flop/B
// -> memory-bound at 23.3 TB/s HBM. CDNA5 data movement:
//   in:  TDM tensor_load_to_lds gathers x[b,c,l0..l0+2) -> xLds[c][b][lt]
//   out: TDM tensor_store_from_lds scatters oLds[b][d][lt] -> out[b,d,l0..l0+2)
// Both are 3D descriptors with dim0 = l (contiguous in HBM); the DMA engine
// owns the 16KB-strided transposes instead of the waves.

typedef float v2f __attribute__((ext_vector_type(2)));
typedef float v4f __attribute__((ext_vector_type(4)));
typedef float v8f __attribute__((ext_vector_type(8)));
typedef unsigned int u32;
typedef unsigned long long u64;
typedef u32 u32x4 __attribute__((ext_vector_type(4)));
typedef u32 u32x8 __attribute__((ext_vector_type(8)));

constexpr int Bsz = 128;
constexpr int Csz = 32;
constexpr int Hsz = 64;
constexpr int Wsz = 64;
constexpr int Lsz = Hsz * Wsz;   // 4096
constexpr int LT  = 2;           // l-positions per workgroup
constexpr int NTHREADS = 256;    // 8 wave32

// Pack a 3D TDM descriptor: dim0 contiguous (LT elems), dim1/dim2 strided.
// ldsA = LDS byte addr, ga = global byte addr, d1/d2 = extents, s1/s2 = strides
// (elements) of dim1/dim2; tensor_dim0 = Lsz for OOB.
__device__ __forceinline__ void tdm_desc(u32x4& g0, u32x8& g1, u32x4& g2, u32x4& g3,
                                         u32 ldsA, u64 ga, u32 d1, u32 s1,
                                         u32 d2, u32 s2) {
  g0[0] = 1u;                                            // count=1, user desc
  g0[1] = ldsA;                                          // lds_addr
  g0[2] = (u32)ga;                                       // global_addr[31:0]
  g0[3] = (u32)((ga >> 32) & 0x01FFFFFFu) | (2u << 30);  // global_addr[56:32], type=2

  g1[0] = (2u << 16);                                    // mask=0, data_size=4B
  g1[1] = ((u32)Lsz & 0xFFFFu) << 16;                    // tensor_dim0[15:0]
  g1[2] = ((u32)Lsz >> 16) | ((d1 & 0xFFFFu) << 16);     // dim0[31:16] | dim1[15:0]
  g1[3] = (d1 >> 16) | ((u32)LT << 16);                  // dim1[31:16] | tile_dim0
  g1[4] = d1 | (d2 << 16);                               // tile_dim1 | tile_dim2
  g1[5] = s1;                                            // dim0_stride[31:0]
  g1[6] = ((s1 >> 16) >> 16) | ((s2 & 0xFFFFu) << 16);   // dim0_stride[47:32] | dim1_stride[15:0]
  g1[7] = (s2 >> 16);                                    // dim1_stride[47:16]

  g2[0] = d2; g2[1] = 0u; g2[2] = 0u; g2[3] = 0u;        // tensor_dim2; dims 3+ unused
  g3[0] = 0u; g3[1] = 0u; g3[2] = 0u; g3[3] = 0u;
}

__global__ __launch_bounds__(NTHREADS)
void dyna_dec_wmma(const float* __restrict__ x,      // [B, C, L]
                   const float* __restrict__ weight, // [L, C, C]
                   const float* __restrict__ bias,   // [L, C]
                   float* __restrict__ out) {        // [B, C, L]
  __shared__ float wLds[LT * Csz * Csz];   // [l][c][d]                    8 KB
  __shared__ float xLds[Csz * Bsz * LT];   // in: [c][b][lt]; out: [b][d][lt] 32 KB

  const int l0  = blockIdx.x * LT;
  const int tid = threadIdx.x;

  // ---- Stage weights: contiguous, coalesced b128 loads ----
  {
    const float* wg = weight + (size_t)l0 * Csz * Csz;
#pragma unroll
    for (int i = 0; i < (LT * Csz * Csz) / (NTHREADS * 4); ++i) {
      const int q = tid + i * NTHREADS;
      v4f v = *(const v4f*)(wg + q * 4);
      *(v4f*)(&wLds[q * 4]) = v;
    }
  }

  // ---- TDM gather: x tile -> xLds[c][b][lt] (wave 0 issues; EXEC ignored) ----
  if (tid < 32) {
    u32x4 g0, g2, g3; u32x8 g1;
    tdm_desc(g0, g1, g2, g3,
             (u32)(uintptr_t)(&xLds[0]),
             (u64)(uintptr_t)(x + (size_t)l0),
             /*d1=*/(u32)Bsz, /*s1=*/(u32)(Csz * Lsz),   // dim1 = b, stride C*L
             /*d2=*/(u32)Csz, /*s2=*/(u32)Lsz);          // dim2 = c, stride L
    asm volatile("tensor_load_to_lds %0, %1, %2, %3"
                 :: "s"(g0), "s"(g1), "s"(g2), "s"(g3) : "memory");
    __builtin_amdgcn_s_wait_tensorcnt(0);
  }
  __syncthreads();

  // ---- Wave work split: 4 waves per l; each wave: one N-half, 4 M-tiles ----
  const int wave    = tid >> 5;
  const int lane    = tid & 31;
  const int ln      = lane & 15;
  const int hi      = lane >> 4;
  const int l_local = wave >> 2;
  const int sub     = wave & 3;
  const int n0      = (sub & 1) * 16;
  const int mBase   = (sub >> 1) * 4;
  const int l_g     = l0 + l_local;

  const float* wl = &wLds[l_local * Csz * Csz];
  const float* xl = &xLds[l_local];          // element stride LT in [c][b][lt]

  // Preload B fragments (8 K-steps of K=4).
  // B 4x16 layout: VGPR0 = rows {0 | 2}, VGPR1 = rows {1 | 3} (low|high half-wave).
  v2f bfrag[8];
#pragma unroll
  for (int k = 0; k < 8; ++k) {
    const int row = 4 * k + hi * 2;
    bfrag[k].x = wl[row * Csz + n0 + ln];
    bfrag[k].y = wl[(row + 1) * Csz + n0 + ln];
  }

  const float biasv = bias[l_g * Csz + n0 + ln];

  v8f acc[4];
#pragma unroll
  for (int j = 0; j < 4; ++j) {
    const int b0 = (mBase + j) * 16;
    acc[j] = (v8f){};
#pragma unroll
    for (int k = 0; k < 8; ++k) {
      // A 16x4: lane<16 -> (M=ln, K=4k+0/1); lane>=16 -> (M=ln, K=4k+2/3)
      const int row = 4 * k + hi * 2;
      v2f a;
      a.x = xl[(row * Bsz + b0 + ln) * LT];
      a.y = xl[((row + 1) * Bsz + b0 + ln) * LT];
      acc[j] = __builtin_amdgcn_wmma_f32_16x16x4_f32(
          /*neg_a=*/false, a, /*neg_b=*/false, bfrag[k],
          /*c_mod=*/(short)0, acc[j], /*reuse_a=*/false, /*reuse_b=*/false);
    }
  }

  // ---- All xLds reads done; reuse it as the output staging tile [b][d][lt] ----
  __syncthreads();
  float* oLds = xLds;
  const int d = n0 + ln;
#pragma unroll
  for (int j = 0; j < 4; ++j) {
    const int brow = (mBase + j) * 16 + hi * 8;
#pragma unroll
    for (int i = 0; i < 8; ++i) {
      // C/D layout: VGPR i holds (M = i + hi*8, N = ln)
      oLds[((brow + i) * Csz + d) * LT + l_local] = acc[j][i] + biasv;
    }
  }
  __syncthreads();

  // ---- TDM scatter: oLds[b][d][lt] -> out[b][d][l0..l0+LT) ----
  // S_ENDPGM's implicit wait-idle drains TENSORcnt before the wave retires.
  if (tid < 32) {
    u32x4 g0, g2, g3; u32x8 g1;
    tdm_desc(g0, g1, g2, g3,
             (u32)(uintptr_t)(&xLds[0]),
             (u64)(uintptr_t)(out + (size_t)l0),
             /*d1=*/(u32)Csz, /*s1=*/(u32)Lsz,           // dim1 = d, stride L
             /*d2=*/(u32)Bsz, /*s2=*/(u32)(Csz * Lsz));  // dim2 = b, stride C*L
    asm volatile("tensor_store_from_lds %0, %1, %2, %3"
                 :: "s"(g0), "s"(g1), "s"(g2), "s"(g3) : "memory");
  }
}

extern "C" void kernel_launch(void* const* d_in, const int* in_sizes, int n_in,
                              void* d_out, int out_size, void* d_ws, size_t ws_size,
                              hipStream_t stream) {
  // setup_inputs order: x, px (unused by reference), weight, bias
  const float* x      = (const float*)d_in[0];
  const float* weight = (const float*)d_in[2];
  const float* bias   = (const float*)d_in[3];
  float* out          = (float*)d_out;

  dim3 grid(Lsz / LT);
  dim3 block(NTHREADS);
  hipLaunchKernelGGL(dyna_dec_wmma, grid, block, 0, stream, x, weight, bias, out);
}